// Speller_87548613362174
// MI455X (gfx1250) — compile-verified
//
#include <hip/hip_runtime.h>
#include <hip/hip_bf16.h>
#include <math.h>

// Problem dims
#define VSZ   5000
#define EDIM  512
#define FDIM  512
#define HDIM  512
#define HEADS 4
#define HD    128
#define STEPS 64
#define BB    32
#define TT    1024
#define LOGITS_LD 5008   // 313 * 16

typedef float v2f __attribute__((ext_vector_type(2)));
typedef float v8f __attribute__((ext_vector_type(8)));

// ---------------------------------------------------------------------------
// Generic M=32 WMMA GEMM: out[32,N] = X[32,K]*W[N,K]^T + bias, optional ReLU.
// One 16x16 output tile per wave, K stepped by 4 via v_wmma_f32_16x16x4_f32.
// ---------------------------------------------------------------------------
__global__ __launch_bounds__(256)
void wmma_mm_kernel(const float* __restrict__ X, const float* __restrict__ W, int K,
                    const float* __restrict__ bias,
                    float* __restrict__ out, int ldo,
                    int Nreal, int nTiles, int totTiles, int doRelu) {
    const int wave = (blockIdx.x * blockDim.x + threadIdx.x) >> 5;
    if (wave >= totTiles) return;                 // whole-wave uniform exit

    const int mTile = wave / nTiles;
    const int nTile = wave - mTile * nTiles;
    const int lane  = threadIdx.x & 31;
    const int l15   = lane & 15;
    const int hi    = lane >> 4;                  // 0 or 1
    const int khalf = hi << 1;                    // K sub-offset 0 or 2

    const int row  = mTile * 16 + l15;            // always < 32 (valid)
    const int ncol = nTile * 16 + l15;
    const int nc   = (ncol < Nreal) ? ncol : 0;   // clamp: garbage cols never stored

    v8f acc = {};
    const float* xr = X + (size_t)row * K + khalf;
    const float* wr = W + (size_t)nc  * K + khalf;
    for (int k = 0; k < K; k += 4) {
        v2f a = *(const v2f*)(xr + k);
        v2f b = *(const v2f*)(wr + k);
        acc = __builtin_amdgcn_wmma_f32_16x16x4_f32(
            /*neg_a=*/false, a, /*neg_b=*/false, b,
            /*c_mod=*/(short)0, acc, /*reuse_a=*/false, /*reuse_b=*/false);
    }

    const float bv = (bias != nullptr) ? bias[nc] : 0.0f;

    if (ncol < Nreal) {
        const int rbase = mTile * 16 + hi * 8;    // C/D layout: M split 0-7 / 8-15
        #pragma unroll
        for (int v = 0; v < 8; ++v) {
            float val = acc[v] + bv;
            if (doRelu) val = fmaxf(val, 0.0f);
            out[(size_t)(rbase + v) * ldo + ncol] = val;
        }
    }
}

// ---------------------------------------------------------------------------
// Fused LSTM step: gates = rnn_in@W_ih^T + h_prev@W_hh^T + (b_ih+b_hh),
// then sigmoid/tanh cell update -- all in one kernel.
// Block = one 16-column slice of H. 8 waves = {4 gates} x {2 M-tiles}.
// Gate tiles staged in LDS, then 256 threads apply the activations.
// h is double-buffered (hprev read by all blocks, hnew written per-slice).
// ---------------------------------------------------------------------------
__global__ __launch_bounds__(256)
void lstm_fused_kernel(const float* __restrict__ X,      // rnn_in [32,1024]
                       const float* __restrict__ hprev,  // [32,512]
                       const float* __restrict__ W_ih,   // [2048,1024]
                       const float* __restrict__ W_hh,   // [2048,512]
                       const float* __restrict__ b_ih,
                       const float* __restrict__ b_hh,
                       float* __restrict__ c,            // [32,512] in/out
                       float* __restrict__ hnew,         // [32,512]
                       float* __restrict__ hcat) {       // [32,1024] (first half)
    __shared__ float s_g[32][65];                 // 32 rows x 64 gate-cols (+pad)

    const int slice = blockIdx.x;                 // 0..31: 16 cols of H
    const int wave  = threadIdx.x >> 5;           // 0..7
    const int gate  = wave >> 1;                  // 0..3 (i,f,g,o)
    const int mTile = wave & 1;
    const int lane  = threadIdx.x & 31;
    const int l15   = lane & 15;
    const int hi    = lane >> 4;
    const int khalf = hi << 1;

    const int row = mTile * 16 + l15;
    const int n   = gate * HDIM + slice * 16 + l15;   // W row, always valid

    v8f acc = {};
    {
        const float* xr = X    + (size_t)row * 1024 + khalf;
        const float* wr = W_ih + (size_t)n   * 1024 + khalf;
        for (int k = 0; k < 1024; k += 4) {
            v2f a = *(const v2f*)(xr + k);
            v2f b = *(const v2f*)(wr + k);
            acc = __builtin_amdgcn_wmma_f32_16x16x4_f32(
                false, a, false, b, (short)0, acc, false, false);
        }
    }
    {
        const float* xr = hprev + (size_t)row * HDIM + khalf;
        const float* wr = W_hh  + (size_t)n   * HDIM + khalf;
        for (int k = 0; k < HDIM; k += 4) {
            v2f a = *(const v2f*)(xr + k);
            v2f b = *(const v2f*)(wr + k);
            acc = __builtin_amdgcn_wmma_f32_16x16x4_f32(
                false, a, false, b, (short)0, acc, false, false);
        }
    }

    const float bv = b_ih[n] + b_hh[n];
    const int rbase = mTile * 16 + hi * 8;
    #pragma unroll
    for (int v = 0; v < 8; ++v)
        s_g[rbase + v][gate * 16 + l15] = acc[v] + bv;
    __syncthreads();

    // activations: 512 (b, j) cells handled by 256 threads
    for (int i = threadIdx.x; i < 512; i += 256) {
        const int b  = i >> 4;
        const int jj = i & 15;
        const int j  = slice * 16 + jj;
        const float gi = s_g[b][jj];
        const float gf = s_g[b][16 + jj];
        const float gg = s_g[b][32 + jj];
        const float go = s_g[b][48 + jj];
        const float si = 1.0f / (1.0f + expf(-gi));
        const float sf = 1.0f / (1.0f + expf(-gf));
        const float so = 1.0f / (1.0f + expf(-go));
        const size_t ci = (size_t)b * HDIM + j;
        const float cn = sf * c[ci] + si * tanhf(gg);
        const float hn = so * tanhf(cn);
        c[ci]    = cn;
        hnew[ci] = hn;
        hcat[(size_t)b * 1024 + j] = hn;
    }
}

// ---------------------------------------------------------------------------
// Step 0 init: rnn_in = concat(embedding[0], listener_feature[:,0,:]); h0=c=0
// ---------------------------------------------------------------------------
__global__ __launch_bounds__(256)
void init_kernel(const float* __restrict__ lf, const float* __restrict__ emb,
                 float* __restrict__ rnn_in, float* __restrict__ h0,
                 float* __restrict__ c) {
    int idx = blockIdx.x * blockDim.x + threadIdx.x;      // 0..32767
    int b = idx >> 10, j = idx & 1023;
    rnn_in[idx] = (j < EDIM) ? emb[j]
                             : lf[(size_t)b * TT * FDIM + (j - EDIM)];
    if (idx < BB * HDIM) { h0[idx] = 0.0f; c[idx] = 0.0f; }
}

// ---------------------------------------------------------------------------
// Attention: one block per (b, head). energy->softmax->context.
// Writes head-3 scores to attn_out; context into ctx and hcat second half.
// ---------------------------------------------------------------------------
__global__ __launch_bounds__(256)
void attn_kernel(const float* __restrict__ h, const float* __restrict__ lf,
                 float* __restrict__ ctx, float* __restrict__ hcat,
                 float* __restrict__ attn_out) {
    __shared__ float s_ds[HD];
    __shared__ float s_e[TT];
    __shared__ float s_red[256];

    const int bh  = blockIdx.x;        // 0..127
    const int b   = bh >> 2;
    const int hh  = bh & 3;
    const int tid = threadIdx.x;

    if (tid < HD) s_ds[tid] = h[(size_t)b * HDIM + hh * HD + tid];
    __syncthreads();

    const float* lfb = lf + (size_t)b * TT * FDIM + hh * HD;

    for (int t = tid; t < TT; t += 256) {
        const float* p = lfb + (size_t)t * FDIM;
        float acc = 0.0f;
        #pragma unroll 4
        for (int d = 0; d < HD; ++d) acc += s_ds[d] * p[d];
        s_e[t] = acc;
    }
    __syncthreads();

    // max reduce
    float m = -1e30f;
    for (int t = tid; t < TT; t += 256) m = fmaxf(m, s_e[t]);
    s_red[tid] = m;
    __syncthreads();
    for (int off = 128; off > 0; off >>= 1) {
        if (tid < off) s_red[tid] = fmaxf(s_red[tid], s_red[tid + off]);
        __syncthreads();
    }
    float mx = s_red[0];
    __syncthreads();

    // exp + sum reduce
    float sum = 0.0f;
    for (int t = tid; t < TT; t += 256) {
        float e = expf(s_e[t] - mx);
        s_e[t] = e;
        sum += e;
    }
    s_red[tid] = sum;
    __syncthreads();
    for (int off = 128; off > 0; off >>= 1) {
        if (tid < off) s_red[tid] += s_red[tid + off];
        __syncthreads();
    }
    float inv = 1.0f / s_red[0];
    for (int t = tid; t < TT; t += 256) s_e[t] *= inv;
    __syncthreads();

    if (hh == HEADS - 1) {
        for (int t = tid; t < TT; t += 256)
            attn_out[(size_t)b * TT + t] = s_e[t];
    }

    // context[d] = sum_t scores[t] * lf[b][t][hh][d]
    for (int d = tid; d < HD; d += 256) {
        float acc = 0.0f;
        for (int t = 0; t < TT; ++t) acc += s_e[t] * lfb[(size_t)t * FDIM + d];
        ctx[(size_t)b * FDIM + hh * HD + d]          = acc;
        hcat[(size_t)b * 1024 + HDIM + hh * HD + d]  = acc;
    }
}

// ---------------------------------------------------------------------------
// Per-batch-row: log-softmax over V=5000 -> raw_pred, argmax -> token,
// assemble next rnn_in = concat(embedding[tok], context).
// ---------------------------------------------------------------------------
__global__ __launch_bounds__(256)
void vocab_kernel(const float* __restrict__ logits, const float* __restrict__ emb,
                  const float* __restrict__ ctx, float* __restrict__ raw_out,
                  float* __restrict__ rnn_in) {
    __shared__ float s_red[256];
    __shared__ int   s_idx[256];
    __shared__ int   s_tok;

    const int b   = blockIdx.x;
    const int tid = threadIdx.x;
    const float* lg = logits + (size_t)b * LOGITS_LD;

    // argmax + max (first-occurrence tie break, matching jnp.argmax)
    float m = -1e30f; int mi = 0;
    for (int v = tid; v < VSZ; v += 256) {
        float x = lg[v];
        if (x > m) { m = x; mi = v; }
    }
    s_red[tid] = m; s_idx[tid] = mi;
    __syncthreads();
    for (int off = 128; off > 0; off >>= 1) {
        if (tid < off) {
            float a = s_red[tid], c2 = s_red[tid + off];
            int   ia = s_idx[tid], ic = s_idx[tid + off];
            if (c2 > a || (c2 == a && ic < ia)) { s_red[tid] = c2; s_idx[tid] = ic; }
        }
        __syncthreads();
    }
    float mx = s_red[0];
    if (tid == 0) s_tok = s_idx[0];
    __syncthreads();

    // sum exp
    float sum = 0.0f;
    for (int v = tid; v < VSZ; v += 256) sum += expf(lg[v] - mx);
    s_red[tid] = sum;
    __syncthreads();
    for (int off = 128; off > 0; off >>= 1) {
        if (tid < off) s_red[tid] += s_red[tid + off];
        __syncthreads();
    }
    float lsum = logf(s_red[0]);

    for (int v = tid; v < VSZ; v += 256)
        raw_out[(size_t)b * VSZ + v] = lg[v] - mx - lsum;

    const int tok = s_tok;
    for (int e = tid; e < EDIM; e += 256)
        rnn_in[(size_t)b * 1024 + e] = emb[(size_t)tok * EDIM + e];
    for (int j = tid; j < FDIM; j += 256)
        rnn_in[(size_t)b * 1024 + EDIM + j] = ctx[(size_t)b * FDIM + j];
}

// ---------------------------------------------------------------------------
extern "C" void kernel_launch(void* const* d_in, const int* in_sizes, int n_in,
                              void* d_out, int out_size, void* d_ws, size_t ws_size,
                              hipStream_t stream) {
    const float* lf   = (const float*)d_in[0];  // [32,1024,512]
    const float* emb  = (const float*)d_in[1];  // [5000,512]
    const float* W_ih = (const float*)d_in[2];  // [2048,1024]
    const float* W_hh = (const float*)d_in[3];  // [2048,512]
    const float* b_ih = (const float*)d_in[4];  // [2048]
    const float* b_hh = (const float*)d_in[5];  // [2048]
    const float* Wp   = (const float*)d_in[6];  // [512,1024]
    const float* bp   = (const float*)d_in[7];  // [512]
    const float* Wc   = (const float*)d_in[8];  // [5000,512]
    const float* bc   = (const float*)d_in[9];  // [5000]

    float* out  = (float*)d_out;
    float* raw  = out;                                   // [64,32,5000]
    float* attn = out + (size_t)STEPS * BB * VSZ;        // [64,32,1024]

    // workspace carve (floats)
    float* ws     = (float*)d_ws;
    float* rnn_in = ws;                    // 32*1024
    float* h0     = rnn_in + BB * 1024;    // 32*512  (double buffer A)
    float* h1     = h0     + BB * HDIM;    // 32*512  (double buffer B)
    float* c      = h1     + BB * HDIM;    // 32*512
    float* hcat   = c      + BB * HDIM;    // 32*1024
    float* ctx    = hcat   + BB * 1024;    // 32*512
    float* hid    = ctx    + BB * FDIM;    // 32*512
    float* logits = hid    + BB * 512;     // 32*5008

    init_kernel<<<128, 256, 0, stream>>>(lf, emb, rnn_in, h0, c);

    for (int s = 0; s < STEPS; ++s) {
        float* hprev = (s & 1) ? h1 : h0;
        float* hnew  = (s & 1) ? h0 : h1;

        // fused: gates GEMM (WMMA) + LSTM cell update
        lstm_fused_kernel<<<32, 256, 0, stream>>>(rnn_in, hprev, W_ih, W_hh,
                                                  b_ih, b_hh, c, hnew, hcat);
        attn_kernel<<<128, 256, 0, stream>>>(hnew, lf, ctx, hcat,
                                             attn + (size_t)s * BB * TT);
        // hid = relu(hcat @ Wp^T + bp)   [32,512]
        wmma_mm_kernel<<<8, 256, 0, stream>>>(hcat, Wp, 1024, bp,
                                              hid, 512, 512, 32, 64, 1);
        // logits = hid @ Wc^T + bc       [32,5000] (padded ld 5008)
        wmma_mm_kernel<<<79, 256, 0, stream>>>(hid, Wc, 512, bc,
                                               logits, LOGITS_LD, VSZ, 313, 626, 0);
        vocab_kernel<<<32, 256, 0, stream>>>(logits, emb, ctx,
                                             raw + (size_t)s * BB * VSZ, rnn_in);
    }
}